// QuantumQLSTM_65481071402812
// MI455X (gfx1250) — compile-verified
//
#include <hip/hip_runtime.h>
#include <hip/hip_bf16.h>
#include <math.h>

typedef __attribute__((ext_vector_type(2))) float v2f;
typedef __attribute__((ext_vector_type(8))) float v8f;

#define T_DIM 512
#define B_DIM 256
#define D_DIM 256
#define H_DIM 256
#define GQ    16   // 4 gates * 4 "qubit" rows

// workspace layout (float offsets, all 16B-aligned)
#define WXP_OFF   0         // [16][256]  x-part of stacked gate weights
#define WHP_OFF   4096      // [16][256]  h-part
#define CROW_OFF  8192      // [256]      final cell row
#define HXBUF_OFF 8448      // [512][256] hx row per timestep
#define XZ_OFF    139520    // [T*B][16]  precomputed x-projections

__device__ __forceinline__ float sigmoidf_(float x) {
  return 1.0f / (1.0f + __expf(-x));
}

// ---------------------------------------------------------------------------
// Kernel 0: pack the four [4,512] gate matrices into Wxp[16][256], Whp[16][256]
// ---------------------------------------------------------------------------
__global__ void pack_weights_kernel(const float* __restrict__ Wf,
                                    const float* __restrict__ Wi,
                                    const float* __restrict__ Wu,
                                    const float* __restrict__ Wo,
                                    float* __restrict__ ws) {
  int idx = blockIdx.x * blockDim.x + threadIdx.x;   // 0 .. 16*512-1
  if (idx >= GQ * 512) return;
  int gq = idx >> 9;          // stacked row: gate*4 + q
  int d  = idx & 511;
  int g = gq >> 2, q = gq & 3;
  const float* W = (g == 0) ? Wf : (g == 1) ? Wi : (g == 2) ? Wu : Wo;
  float v = W[q * 512 + d];
  if (d < D_DIM) ws[WXP_OFF + gq * D_DIM + d] = v;
  else           ws[WHP_OFF + gq * H_DIM + (d - D_DIM)] = v;
}

// ---------------------------------------------------------------------------
// Kernel 1: XZ[n][gq] = X[n][:] . Wxp[gq][:]   (n = t*B + b, 131072 rows)
// One wave per 16-row M tile; N=16 is exactly one WMMA tile; K-loop of 64
// V_WMMA_F32_16X16X4_F32 ops. fp32 layout per ISA 7.12.2:
//   A 16x4 : lanes0-15 M=0..15 hold K=k,k+1 in v0,v1; lanes16-31 hold K=k+2,k+3
//   B 4x16 : mirrored with lane -> N
//   C/D    : vgpr r -> M=r (lanes0-15) / M=r+8 (lanes16-31), lane&15 -> N
// ---------------------------------------------------------------------------
__global__ void __launch_bounds__(256) xz_gemm_kernel(const float* __restrict__ X,
                                                      float* __restrict__ ws) {
  const float* Wxp = ws + WXP_OFF;
  float*       XZ  = ws + XZ_OFF;

  int lane = threadIdx.x & 31;
  int wave = threadIdx.x >> 5;
  int tile = blockIdx.x * 8 + wave;       // 8192 tiles * 16 rows = 131072 = T*B
  int n0   = tile * 16;
  int hi   = lane >> 4;                   // 0 or 1 -> K offset 0 / 2
  int lo   = lane & 15;                   // M (for A) / N (for B)
  int koff = 2 * hi;

  const float* Arow = X   + (size_t)(n0 + lo) * D_DIM;
  const float* Brow = Wxp + (size_t)lo * D_DIM;

  // warm the A row into cache (emits global_prefetch_b8)
  #pragma unroll
  for (int pk = 0; pk < D_DIM; pk += 64)
    __builtin_prefetch(Arow + pk, 0, 3);

  v8f acc = {};
  #pragma unroll 4
  for (int k = 0; k < D_DIM; k += 4) {
    v2f a = *(const v2f*)(Arow + k + koff);
    v2f b = *(const v2f*)(Brow + k + koff);
    acc = __builtin_amdgcn_wmma_f32_16x16x4_f32(
        /*neg_a=*/false, a, /*neg_b=*/false, b,
        /*c_mod=*/(short)0, acc, /*reuse_a=*/false, /*reuse_b=*/false);
  }

  union { v8f v; float f[8]; } u;
  u.v = acc;
  #pragma unroll
  for (int r = 0; r < 8; ++r)
    XZ[(size_t)(n0 + r + 8 * hi) * GQ + lo] = u.f[r];
}

// ---------------------------------------------------------------------------
// Kernel 2: sequential scan over T=512 steps, one block of 256 threads.
// Thread h owns c[h], hx[h]. Per step:
//   hz[gq] = sum_h hx[h]*Whp[gq][h]   (shfl_xor wave reduce + LDS cross-wave)
//   z[gq]  = XZ[t*B + h][gq] + hz[gq] + (bias[q] + phi[q])
//   gate_g = act( prod_{q=1..3} cos(z[g*4+q]) )
//   c = f*c + i*u ; hx = o*tanh(c) ; hxbuf[t][h] = hx
// ---------------------------------------------------------------------------
__global__ void __launch_bounds__(256) qlstm_scan_kernel(
    float* __restrict__ ws,
    const float* __restrict__ bf, const float* __restrict__ pf,
    const float* __restrict__ bi, const float* __restrict__ pi,
    const float* __restrict__ bu, const float* __restrict__ pu,
    const float* __restrict__ bo, const float* __restrict__ po) {
  __shared__ float red[8 * GQ];
  __shared__ float hz_sh[GQ];

  int tid  = threadIdx.x;          // == h == b (H == B trick)
  int lane = tid & 31;
  int wid  = tid >> 5;

  const float* Whp  = ws + WHP_OFF;
  const float* XZ   = ws + XZ_OFF;
  float* hxbuf = ws + HXBUF_OFF;
  float* crow  = ws + CROW_OFF;

  float wh[GQ];
  #pragma unroll
  for (int gq = 0; gq < GQ; ++gq) wh[gq] = Whp[gq * H_DIM + tid];

  float ac[GQ];  // bias + phi folded into the cos() argument
  #pragma unroll
  for (int q = 0; q < 4; ++q) {
    ac[0 * 4 + q] = bf[q] + pf[q];
    ac[1 * 4 + q] = bi[q] + pi[q];
    ac[2 * 4 + q] = bu[q] + pu[q];
    ac[3 * 4 + q] = bo[q] + po[q];
  }

  float c = 0.0f, hx = 0.0f;

  for (int t = 0; t < T_DIM; ++t) {
    // --- hz[gq] = <hx, Whp[gq]> over 256 threads ---
    #pragma unroll
    for (int gq = 0; gq < GQ; ++gq) {
      float p = hx * wh[gq];
      #pragma unroll
      for (int m = 16; m >= 1; m >>= 1) p += __shfl_xor(p, m, 32);
      if (lane == 0) red[wid * GQ + gq] = p;
    }
    __syncthreads();
    if (tid < GQ) {
      float s = 0.0f;
      #pragma unroll
      for (int w = 0; w < 8; ++w) s += red[w * GQ + tid];
      hz_sh[tid] = s;
    }
    __syncthreads();

    // --- gates for batch element b == tid ---
    const float* xz = XZ + ((size_t)t * B_DIM + tid) * GQ;
    float z[GQ];
    #pragma unroll
    for (int gq = 0; gq < GQ; ++gq) z[gq] = xz[gq] + hz_sh[gq] + ac[gq];

    float pF = __cosf(z[1])  * __cosf(z[2])  * __cosf(z[3]);
    float pI = __cosf(z[5])  * __cosf(z[6])  * __cosf(z[7]);
    float pU = __cosf(z[9])  * __cosf(z[10]) * __cosf(z[11]);
    float pO = __cosf(z[13]) * __cosf(z[14]) * __cosf(z[15]);

    float f = sigmoidf_(pF);
    float i = sigmoidf_(pI);
    float u = tanhf(pU);
    float o = sigmoidf_(pO);

    c  = f * c + i * u;
    hx = o * tanhf(c);
    hxbuf[t * H_DIM + tid] = hx;
    // no extra barrier needed: next iteration's red[] writes occur only after
    // this iteration's second __syncthreads(), which follows all red[] reads.
  }
  crow[tid] = c;
}

// ---------------------------------------------------------------------------
// Kernel 3: broadcast hx rows into the 128 MB output + final hx/cx tails.
// out = [ outs (T*B*H) | hx (B*H) | cx (B*H) ], all rows constant across b.
// float4 stores; hxbuf (512 KB) stays L2-resident.
// ---------------------------------------------------------------------------
__global__ void __launch_bounds__(256) bcast_out_kernel(const float* __restrict__ ws,
                                                        float4* __restrict__ out) {
  const size_t OUTS_V4 = (size_t)T_DIM * B_DIM * H_DIM / 4;  // 8388608
  const size_t BH_V4   = (size_t)B_DIM * H_DIM / 4;          // 16384
  const int    HV4     = H_DIM / 4;                          // 64 (pow2)

  size_t idx = (size_t)blockIdx.x * blockDim.x + threadIdx.x;
  if (idx >= OUTS_V4 + 2 * BH_V4) return;

  const float4* hxbuf4 = (const float4*)(ws + HXBUF_OFF);
  const float4* crow4  = (const float4*)(ws + CROW_OFF);
  int h4 = (int)(idx & (HV4 - 1));   // valid for every region (64 | 16384)

  float4 v;
  if (idx < OUTS_V4) {
    size_t t = idx / BH_V4;
    v = hxbuf4[t * HV4 + h4];
  } else if (idx < OUTS_V4 + BH_V4) {
    v = hxbuf4[(size_t)(T_DIM - 1) * HV4 + h4];   // final hx
  } else {
    v = crow4[h4];                                // final cx
  }
  out[idx] = v;
}

// ---------------------------------------------------------------------------
extern "C" void kernel_launch(void* const* d_in, const int* in_sizes, int n_in,
                              void* d_out, int out_size, void* d_ws, size_t ws_size,
                              hipStream_t stream) {
  const float* X  = (const float*)d_in[0];
  const float* Wf = (const float*)d_in[1];
  const float* bf = (const float*)d_in[2];
  const float* pf = (const float*)d_in[3];
  const float* Wi = (const float*)d_in[4];
  const float* bi = (const float*)d_in[5];
  const float* pi = (const float*)d_in[6];
  const float* Wu = (const float*)d_in[7];
  const float* bu = (const float*)d_in[8];
  const float* pu = (const float*)d_in[9];
  const float* Wo = (const float*)d_in[10];
  const float* bo = (const float*)d_in[11];
  const float* po = (const float*)d_in[12];
  float* ws = (float*)d_ws;

  // 0) pack stacked gate weights
  pack_weights_kernel<<<32, 256, 0, stream>>>(Wf, Wi, Wu, Wo, ws);

  // 1) WMMA GEMM: XZ[t*B+b][16] = x . Wx^T   (8192 wave-tiles, 8 waves/block)
  xz_gemm_kernel<<<1024, 256, 0, stream>>>(X, ws);

  // 2) sequential recurrence (single workgroup, 512 steps)
  qlstm_scan_kernel<<<1, 256, 0, stream>>>(ws, bf, pf, bi, pi, bu, pu, bo, po);

  // 3) broadcast to full output
  size_t total_v4 = (size_t)T_DIM * B_DIM * H_DIM / 4 + 2 * ((size_t)B_DIM * H_DIM / 4);
  int blocks = (int)((total_v4 + 255) / 256);
  bcast_out_kernel<<<blocks, 256, 0, stream>>>(ws, (float4*)d_out);
}